// DecayEnvelopeGenerator_59176059404846
// MI455X (gfx1250) — compile-verified
//
#include <hip/hip_runtime.h>

typedef __attribute__((ext_vector_type(2))) float v2f;
typedef __attribute__((ext_vector_type(8))) float v8f;

#define N_SAMP  32000
#define NFRAME  63
#define LOG2E   1.4426950408889634f

// out[bp][s] = max_f ( s >= 512*f ? scale_f * exp(-100*d_f*(s-512f)/N) : 0 )
// One wave handles a 16-sample strip; each V_WMMA_F32_16X16X4_F32 produces a
// 16-frame x 16-sample tile of exp2 arguments (rank-2 outer product).
// Frame-tiles that are entirely in the future of this strip (sbase < 512*fbase)
// are skipped with a wave-uniform branch (EXEC stays all-1s for the WMMAs).
__global__ __launch_bounds__(256) void decay_env_wmma(const float* __restrict__ dparam,
                                                      float* __restrict__ out) {
  __shared__ float s_scale[64];
  __shared__ float s_c[64];   // c_f = -100*d_f/N * log2(e)  (exp2 coefficient)

  const int bp  = blockIdx.y;      // 0..23  (batch*pitch)
  const int tid = threadIdx.x;

  if (tid < 64) {
    const float d = (tid < NFRAME) ? dparam[bp * NFRAME + tid] : 0.0f;
    const float a100  = -100.0f * d;
    const float t_max = (float)(N_SAMP - 1) / (float)N_SAMP;
    // inf-norm of d*exp(-100 d t) over t in [0, t_max]
    const float nrm   = fabsf(d) * __builtin_amdgcn_exp2f(fmaxf(a100, 0.0f) * (t_max * LOG2E));
    const float scale = d / fmaxf(nrm, 1e-12f);
    s_scale[tid] = (tid < NFRAME) ? scale : 0.0f;
    s_c[tid]     = (tid < NFRAME) ? (a100 * (LOG2E / (float)N_SAMP)) : 0.0f;
  }
  __syncthreads();

  const int lane = tid & 31;
  const int wave = tid >> 5;
  const int hi   = lane >> 4;          // 0: lanes 0-15 (K=0,1 / M rows 0-7), 1: lanes 16-31
  const int nl   = lane & 15;          // A row index == B/D column index
  const int sbase = blockIdx.x * 128 + wave * 16;   // wave-uniform, 16-aligned
  const float s_n = (float)(sbase + nl);

  float macc[8];
#pragma unroll
  for (int v = 0; v < 8; ++v) macc[v] = 0.0f;   // invalid frames contribute 0, like ref

  const v8f czero = {};
#pragma unroll
  for (int ft = 0; ft < 4; ++ft) {
    const int fbase = ft * 16;
    // Wave-uniform liveness: tile has any valid frame iff sbase >= 512*fbase
    // (sbase is 16-aligned, 512*fbase is 512-aligned, so this is exact).
    if (sbase >= (fbase << 9)) {
      // A (16x4 f32): lane holds row m=nl; lanes0-15 = K0/K1, lanes16-31 = K2/K3 (zeroed).
      const float cm = hi ? 0.0f : s_c[fbase + nl];
      v2f a;
      a.x = cm;
      a.y = cm * (-512.0f * (float)(fbase + nl));
      // B (4x16 f32): lane holds col n=nl; K=0 row = s_n, K=1 row = 1; upper half zeroed.
      v2f b;
      b.x = hi ? 0.0f : s_n;
      b.y = hi ? 0.0f : 1.0f;
      // arg[m][n] = c_m * (s_n - 512*f_m)   (one WMMA = 16x16 argument tile)
      v8f arg = __builtin_amdgcn_wmma_f32_16x16x4_f32(
          /*neg_a=*/false, a, /*neg_b=*/false, b,
          /*c_mod=*/(short)0, czero, /*reuse_a=*/false, /*reuse_b=*/false);
#pragma unroll
      for (int v = 0; v < 8; ++v) {
        const int m     = fbase + v + (hi << 3);        // frame index of this D element
        const int t_rel = (sbase + nl) - (m << 9);      // validity: s >= 512*f
        float env = s_scale[m] * __builtin_amdgcn_exp2f(arg[v]);
        env = (t_rel >= 0) ? env : 0.0f;
        macc[v] = fmaxf(macc[v], env);
      }
    }
  }

  // reduce over the 8 M-rows held per lane, then across lane halves (wave32 shuffle)
  float pm = macc[0];
#pragma unroll
  for (int v = 1; v < 8; ++v) pm = fmaxf(pm, macc[v]);
  pm = fmaxf(pm, __shfl_xor(pm, 16, 32));

  if (hi == 0) {
    out[bp * N_SAMP + sbase + nl] = pm;
  }
}

extern "C" void kernel_launch(void* const* d_in, const int* in_sizes, int n_in,
                              void* d_out, int out_size, void* d_ws, size_t ws_size,
                              hipStream_t stream) {
  (void)in_sizes; (void)n_in; (void)out_size; (void)d_ws; (void)ws_size;
  const float* dparam = (const float*)d_in[0];   // (4, 6, 63) f32
  float* out = (float*)d_out;                    // (4, 6, 32000) f32
  dim3 grid(N_SAMP / 128, 24, 1);                // 8 waves/block * 16 samples/wave
  decay_env_wmma<<<grid, dim3(256, 1, 1), 0, stream>>>(dparam, out);
}